// AffineExponential_28698971472502
// MI455X (gfx1250) — compile-verified
//
#include <hip/hip_runtime.h>

typedef __attribute__((ext_vector_type(2))) float v2f;
typedef __attribute__((ext_vector_type(8))) float v8f;

#define DD 128
#define VSTR 132                  // padded V-buffer stride (floats): bank-conflict-free
#define WAVES_PER_BLOCK 4
#define ROWS_PER_WAVE 16
#define ROWS_PER_BLOCK (WAVES_PER_BLOCK * ROWS_PER_WAVE)   // 64
#define NTERMS 24

// 1/k, correctly rounded at compile time (avoids per-term v_div_* sequence)
__device__ __constant__ float INVK[NTERMS + 1] = {
    0.0f,        1.0f,        1.0f/2.0f,  1.0f/3.0f,  1.0f/4.0f,  1.0f/5.0f,
    1.0f/6.0f,   1.0f/7.0f,   1.0f/8.0f,  1.0f/9.0f,  1.0f/10.0f, 1.0f/11.0f,
    1.0f/12.0f,  1.0f/13.0f,  1.0f/14.0f, 1.0f/15.0f, 1.0f/16.0f, 1.0f/17.0f,
    1.0f/18.0f,  1.0f/19.0f,  1.0f/20.0f, 1.0f/21.0f, 1.0f/22.0f, 1.0f/23.0f,
    1.0f/24.0f
};

// y = expm(t*W) x + bias*t  via Taylor recurrence v_k = (t/k) W v_{k-1}
// One wave = one 16-sample tile. V' = V * W^T done with V_WMMA_F32_16X16X4_F32.
__global__ __launch_bounds__(128)
void affine_expm_kernel(const float* __restrict__ X,
                        const float* __restrict__ T,
                        const float* __restrict__ W,
                        const float* __restrict__ Bias,
                        float* __restrict__ Yout,
                        float* __restrict__ LJD)
{
    // W^T in LDS: sWT[j*128 + i] = W[i][j].  B-frag gather (column across lanes)
    // is bank-conflict-free for stride 128.
    __shared__ __align__(16) float sWT[DD * DD];                 // 64 KB
    __shared__ __align__(16) float sV[ROWS_PER_BLOCK * VSTR];    // 33 KB
    __shared__ __align__(16) float sBias[DD];
    __shared__ __align__(16) float sT[ROWS_PER_BLOCK];

    const int tid  = threadIdx.x;
    const int wave = tid >> 5;
    const int lane = tid & 31;
    const int lo16 = lane & 15;
    const int hi   = lane >> 4;                 // 0: K-even pair / M rows 0-7 ; 1: +2 / rows 8-15

    const int blockRow = blockIdx.x * ROWS_PER_BLOCK;
    const int waveRow  = blockRow + wave * ROWS_PER_WAVE;

    // ---- stage W transposed into LDS (128 elements per thread)
    for (int idx = tid; idx < DD * DD; idx += blockDim.x) {
        int i = idx >> 7;
        int j = idx & (DD - 1);
        sWT[j * DD + i] = W[idx];
    }
    if (tid < DD) sBias[tid] = Bias[tid];
    if (tid < ROWS_PER_BLOCK) sT[tid] = T[blockRow + tid];

    // ---- stage this wave's 16x128 X tile into its V buffer (row-major, stride VSTR)
    float* vbuf = &sV[wave * ROWS_PER_WAVE * VSTR];
    {
        const float4* Xv = (const float4*)(X + (size_t)waveRow * DD);
        for (int q = lane; q < ROWS_PER_WAVE * (DD / 4); q += 32) {
            int r  = q >> 5;              // 32 float4 per row
            int c4 = q & 31;
            float4 v = Xv[r * 32 + c4];
            float* p = &vbuf[r * VSTR + c4 * 4];
            p[0] = v.x; p[1] = v.y; p[2] = v.z; p[3] = v.w;
        }
    }
    __syncthreads();

    // per-lane t values for the C/D rows this lane holds (M = hi*8 + i)
    float tv[8];
    #pragma unroll
    for (int i = 0; i < 8; ++i) tv[i] = sT[wave * 16 + hi * 8 + i];

    // Y accumulator init = x (k=0 term), read from vbuf in C/D layout
    v8f Yacc[8];
    #pragma unroll
    for (int tn = 0; tn < 8; ++tn) {
        int ncol = tn * 16 + lo16;
        #pragma unroll
        for (int i = 0; i < 8; ++i)
            Yacc[tn][i] = vbuf[(hi * 8 + i) * VSTR + ncol];
    }

    // ---- Taylor recurrence
    for (int k = 1; k <= NTERMS; ++k) {
        float invk = INVK[k];
        v8f nv[8];
        #pragma unroll
        for (int tn = 0; tn < 8; ++tn) nv[tn] = (v8f){0,0,0,0,0,0,0,0};

        // V' tiles = V(16x128) * W^T(128x128), K swept in chunks of 4 (fp32 WMMA)
        #pragma unroll 4
        for (int kk = 0; kk < 32; ++kk) {
            int kbase = kk * 4 + hi * 2;
            // A-frag: row = lo16, cols kbase..kbase+1  (16x4 f32 layout)
            v2f a = *(const v2f*)&vbuf[lo16 * VSTR + kbase];
            #pragma unroll
            for (int tn = 0; tn < 8; ++tn) {
                int ncol = tn * 16 + lo16;
                // B-frag: B[j][n] = W[ncol][j] = sWT[j*128 + ncol]
                v2f b;
                b.x = sWT[(kbase + 0) * DD + ncol];
                b.y = sWT[(kbase + 1) * DD + ncol];
                nv[tn] = __builtin_amdgcn_wmma_f32_16x16x4_f32(
                    false, a, false, b, (short)0, nv[tn], false, false);
            }
        }

        // scale by t/k, accumulate into Y, write V' back (C/D layout, conflict-free).
        // All LDS reads above precede these stores in wave program order -> safe.
        #pragma unroll
        for (int tn = 0; tn < 8; ++tn) {
            int ncol = tn * 16 + lo16;
            #pragma unroll
            for (int i = 0; i < 8; ++i) {
                float val = nv[tn][i] * (tv[i] * invk);
                Yacc[tn][i] += val;
                vbuf[(hi * 8 + i) * VSTR + ncol] = val;
            }
        }
    }

    // ---- epilogue: y += bias*t ; emit y and ljd = diag(W)*t
    #pragma unroll
    for (int tn = 0; tn < 8; ++tn) {
        int ncol  = tn * 16 + lo16;
        float bcol  = sBias[ncol];
        float wdiag = sWT[ncol * DD + ncol];
        #pragma unroll
        for (int i = 0; i < 8; ++i) {
            int row = waveRow + hi * 8 + i;
            Yout[(size_t)row * DD + ncol] = Yacc[tn][i] + bcol * tv[i];
            LJD [(size_t)row * DD + ncol] = wdiag * tv[i];
        }
    }
}

extern "C" void kernel_launch(void* const* d_in, const int* in_sizes, int n_in,
                              void* d_out, int out_size, void* d_ws, size_t ws_size,
                              hipStream_t stream) {
    const float* X    = (const float*)d_in[0];   // [B,128]
    const float* T    = (const float*)d_in[1];   // [B,1]
    const float* W    = (const float*)d_in[2];   // [128,128]
    const float* Bias = (const float*)d_in[3];   // [1,128]

    const int B = in_sizes[1];                   // t has B elements
    float* Y   = (float*)d_out;                  // first B*128 floats
    float* LJD = (float*)d_out + (size_t)B * DD; // next  B*128 floats

    dim3 grid(B / ROWS_PER_BLOCK);
    dim3 block(WAVES_PER_BLOCK * 32);
    affine_expm_kernel<<<grid, block, 0, stream>>>(X, T, W, Bias, Y, LJD);
}